// GroupedQueryAttention_40157944218020
// MI455X (gfx1250) — compile-verified
//
#include <hip/hip_runtime.h>
#include <math.h>

// ---------------------------------------------------------------------------
// Grouped Query Attention for gfx1250 (MI455X), fp32 WMMA pipeline.
//   B=2, S=2048, D=2048, H=16, KV=4, dk=128, G=4
// Stages (all GEMMs via V_WMMA_F32_16X16X4_F32):
//   1) Q = query @ Wq^T        [4096 x 2048]
//   2) K = key   @ Wk^T        [4096 x  512]
//   3) V = value @ Wv^T        [4096 x  512]
//   4) scores[b,h] = Q_bh @ K_bkv^T  -> attn region of d_out (raw)
//   5) softmax rows (applies 1/sqrt(dk) scale) in place
//   6) ctx[b,h] = attn_bh @ V_bkv    -> ws
//   7) out = ctx @ Wo^T              -> d_out
//
// LDS tiles are stored K-pair-interleaved: element (k, m) at
//   tile[k>>1][m].{x|y}  (x for even k, y for odd k)
// so each WMMA fragment (two K values per lane) is ONE aligned ds_load_b64
// directly into the even-aligned VGPR pair -- no repack moves.
// Row stride 288 floats (== 32 mod 64 banks) makes the two lane-halves
// (K-pair rows r and r+1) hit disjoint bank windows: conflict-free reads.
// ---------------------------------------------------------------------------

typedef float v2f __attribute__((ext_vector_type(2)));
typedef float v8f __attribute__((ext_vector_type(8)));

static constexpr int BM  = 128;           // block tile M
static constexpr int BN  = 128;           // block tile N
static constexpr int BK  = 32;            // block tile K (16 K-pair rows x 2)
static constexpr int RS2 = 2 * BM + 32;   // floats per K-pair row (288)

// Generic batched GEMM:  C = A * B   (or A * B^T when BT=true)
//   A: [M x K] row-major (lda)
//   B: BT ? [N x K] row-major (ldb)  : [K x N] row-major (ldb)
//   C: [M x N] row-major (ldc)
// Batch z: outer = z / innerCount, inner = z % innerCount
//   off = outer*s?o + (inner / g?) * s?i      (per A/B/C)
template <bool BT>
__global__ __launch_bounds__(256) void wmma_gemm_f32(
    const float* __restrict__ Ag, const float* __restrict__ Bg,
    float* __restrict__ Cg, int M, int N, int K, int lda, int ldb, int ldc,
    int innerCount,
    long long sAo, long long sAi, int gA,
    long long sBo, long long sBi, int gB,
    long long sCo, long long sCi, int gC) {
  const int tid  = threadIdx.x;
  const int lane = tid & 31;
  const int w    = tid >> 5;      // wave 0..7
  const int wm   = w >> 2;        // 0..1  -> 64-row strip
  const int wn   = w & 3;         // 0..3  -> 32-col strip
  const int m0   = blockIdx.y * BM;
  const int n0   = blockIdx.x * BN;

  const int z     = blockIdx.z;
  const int outer = z / innerCount;
  const int inner = z - outer * innerCount;
  const float* A = Ag + (outer * sAo + (long long)(inner / gA) * sAi);
  const float* B = Bg + (outer * sBo + (long long)(inner / gB) * sBi);
  float*       C = Cg + (outer * sCo + (long long)(inner / gC) * sCi);

  // K-pair-interleaved tiles: row r holds K = {2r, 2r+1}, column m as float2.
  __shared__ float As[(BK / 2) * RS2];   // 16 * 288 * 4B = 18.4 KB
  __shared__ float Bs[(BK / 2) * RS2];

  v8f acc[4][2];
#pragma unroll
  for (int mt = 0; mt < 4; ++mt)
#pragma unroll
    for (int nt = 0; nt < 2; ++nt)
      acc[mt][nt] = (v8f){0.f, 0.f, 0.f, 0.f, 0.f, 0.f, 0.f, 0.f};

  // ---- staging thread maps (float4 global reads) ----
  // A (and B when BT): read along K. 128 rows x 32 K = 4096 floats.
  const int sm  = tid >> 3;          // row stripe 0..31 (4 passes of +32)
  const int sk4 = (tid & 7) * 4;     // K offset 0,4,...,28
  // B when !BT: read along N, pairs of K rows. 32 K x 128 N = 4096 floats.
  const int sn4 = (tid & 31) * 4;    // N offset
  const int sk2 = tid >> 5;          // K-pair row 0..7 (2 passes of +8)

  const int r16 = lane & 15;         // fragment row/col within 16
  const int kh  = lane >> 4;         // lane half: K-pair row offset 0/1

  for (int k0 = 0; k0 < K; k0 += BK) {
    // ---- stage A: As[k>>1][m] = {A[m][k], A[m][k+1]}
#pragma unroll
    for (int p = 0; p < 4; ++p) {
      const int m = sm + p * 32;
      const float4 a4 =
          *(const float4*)&A[(long long)(m0 + m) * lda + (k0 + sk4)];
      *(v2f*)&As[((sk4 >> 1) + 0) * RS2 + 2 * m] = (v2f){a4.x, a4.y};
      *(v2f*)&As[((sk4 >> 1) + 1) * RS2 + 2 * m] = (v2f){a4.z, a4.w};
    }
    // ---- stage B: Bs[k>>1][n] = {B[k][n], B[k+1][n]}
    if (BT) {
#pragma unroll
      for (int p = 0; p < 4; ++p) {
        const int n = sm + p * 32;
        const float4 b4 =
            *(const float4*)&B[(long long)(n0 + n) * ldb + (k0 + sk4)];
        *(v2f*)&Bs[((sk4 >> 1) + 0) * RS2 + 2 * n] = (v2f){b4.x, b4.y};
        *(v2f*)&Bs[((sk4 >> 1) + 1) * RS2 + 2 * n] = (v2f){b4.z, b4.w};
      }
    } else {
#pragma unroll
      for (int p = 0; p < 2; ++p) {
        const int r = sk2 + p * 8;   // K-pair row 0..15
        const float4 b0 =
            *(const float4*)&B[(long long)(k0 + 2 * r + 0) * ldb + (n0 + sn4)];
        const float4 b1 =
            *(const float4*)&B[(long long)(k0 + 2 * r + 1) * ldb + (n0 + sn4)];
        float* brow = &Bs[r * RS2];
        *(v2f*)&brow[2 * (sn4 + 0)] = (v2f){b0.x, b1.x};
        *(v2f*)&brow[2 * (sn4 + 1)] = (v2f){b0.y, b1.y};
        *(v2f*)&brow[2 * (sn4 + 2)] = (v2f){b0.z, b1.z};
        *(v2f*)&brow[2 * (sn4 + 3)] = (v2f){b0.w, b1.w};
      }
    }
    __syncthreads();

    // ---- compute: 8 WMMA K-steps of 4 within the BK=32 slab
#pragma unroll
    for (int kk = 0; kk < BK; kk += 4) {
      const int r = (kk >> 1) + kh;  // this lane-half's K-pair row
      const float* arow = &As[r * RS2];
      const float* brow = &Bs[r * RS2];
      v2f af[4], bf[2];
#pragma unroll
      for (int mt = 0; mt < 4; ++mt)
        af[mt] = *(const v2f*)&arow[2 * (wm * 64 + mt * 16 + r16)];
#pragma unroll
      for (int nt = 0; nt < 2; ++nt)
        bf[nt] = *(const v2f*)&brow[2 * (wn * 32 + nt * 16 + r16)];
#pragma unroll
      for (int mt = 0; mt < 4; ++mt)
#pragma unroll
        for (int nt = 0; nt < 2; ++nt)
          acc[mt][nt] = __builtin_amdgcn_wmma_f32_16x16x4_f32(
              false, af[mt], false, bf[nt], (short)0, acc[mt][nt], false,
              false);
    }
    __syncthreads();
  }

  // ---- write back (16x16 f32 C layout: lane half -> M offset 0/8, VGPR r -> M row)
  const int col = lane & 15;
  const int mb  = (lane >> 4) * 8;
#pragma unroll
  for (int mt = 0; mt < 4; ++mt)
#pragma unroll
    for (int nt = 0; nt < 2; ++nt)
#pragma unroll
      for (int r = 0; r < 8; ++r) {
        const int mm = m0 + wm * 64 + mt * 16 + mb + r;
        const int nn = n0 + wn * 32 + nt * 16 + col;
        C[(long long)mm * ldc + nn] = acc[mt][nt][r];
      }
}

// Row softmax over 2048 columns, applying `scale` before exp (softmax(scale*x)).
__global__ __launch_bounds__(256) void softmax_rows_2048(float* __restrict__ attn,
                                                         float scale) {
  const long long row = blockIdx.x;
  float4* p = (float4*)(attn + row * 2048ll);
  __shared__ float red[256];
  const int tid = threadIdx.x;

  float4 v0 = p[tid * 2 + 0];
  float4 v1 = p[tid * 2 + 1];
  float m = fmaxf(fmaxf(fmaxf(v0.x, v0.y), fmaxf(v0.z, v0.w)),
                  fmaxf(fmaxf(v1.x, v1.y), fmaxf(v1.z, v1.w)));
  red[tid] = m;
  __syncthreads();
  for (int s = 128; s > 0; s >>= 1) {
    if (tid < s) red[tid] = fmaxf(red[tid], red[tid + s]);
    __syncthreads();
  }
  m = red[0];
  __syncthreads();

  v0.x = __expf(scale * (v0.x - m)); v0.y = __expf(scale * (v0.y - m));
  v0.z = __expf(scale * (v0.z - m)); v0.w = __expf(scale * (v0.w - m));
  v1.x = __expf(scale * (v1.x - m)); v1.y = __expf(scale * (v1.y - m));
  v1.z = __expf(scale * (v1.z - m)); v1.w = __expf(scale * (v1.w - m));
  const float sum = (v0.x + v0.y + v0.z + v0.w) + (v1.x + v1.y + v1.z + v1.w);
  red[tid] = sum;
  __syncthreads();
  for (int s = 128; s > 0; s >>= 1) {
    if (tid < s) red[tid] += red[tid + s];
    __syncthreads();
  }
  const float inv = 1.f / red[0];
  v0.x *= inv; v0.y *= inv; v0.z *= inv; v0.w *= inv;
  v1.x *= inv; v1.y *= inv; v1.z *= inv; v1.w *= inv;
  p[tid * 2 + 0] = v0;
  p[tid * 2 + 1] = v1;
}

extern "C" void kernel_launch(void* const* d_in, const int* in_sizes, int n_in,
                              void* d_out, int out_size, void* d_ws, size_t ws_size,
                              hipStream_t stream) {
  (void)in_sizes; (void)n_in; (void)out_size; (void)ws_size;
  constexpr int B = 2, S = 2048, D = 2048, H = 16, KV = 4, DK = 128;
  constexpr long long BS = (long long)B * S;            // 4096
  const float scale = 0.08838834764831845f;             // 1/sqrt(128)

  const float* q  = (const float*)d_in[0];
  const float* k  = (const float*)d_in[1];
  const float* v  = (const float*)d_in[2];
  const float* Wq = (const float*)d_in[3];
  const float* Wk = (const float*)d_in[4];
  const float* Wv = (const float*)d_in[5];
  const float* Wo = (const float*)d_in[6];

  float* out  = (float*)d_out;                          // [B*S, D]
  float* attn = out + BS * D;                           // [B, H, S, S]

  float* Qp = (float*)d_ws;                             // [B*S, D]     32 MB
  float* Kp = Qp + BS * D;                              // [B*S, KV*DK]  8 MB
  float* Vp = Kp + BS * (KV * DK);                      // [B*S, KV*DK]  8 MB
  float* Cx = Vp + BS * (KV * DK);                      // [B*S, D]     32 MB

  dim3 blk(256);

  // 1-3) projections: Y = X @ W^T  (BT = true), single batch
  wmma_gemm_f32<true><<<dim3(D / BN, BS / BM, 1), blk, 0, stream>>>(
      q, Wq, Qp, (int)BS, D, D, D, D, D, 1, 0, 0, 1, 0, 0, 1, 0, 0, 1);
  wmma_gemm_f32<true><<<dim3((KV * DK) / BN, BS / BM, 1), blk, 0, stream>>>(
      k, Wk, Kp, (int)BS, KV * DK, D, D, D, KV * DK, 1, 0, 0, 1, 0, 0, 1, 0, 0, 1);
  wmma_gemm_f32<true><<<dim3((KV * DK) / BN, BS / BM, 1), blk, 0, stream>>>(
      v, Wv, Vp, (int)BS, KV * DK, D, D, D, KV * DK, 1, 0, 0, 1, 0, 0, 1, 0, 0, 1);

  // 4) scores[b,h] = Q_bh [S,dk] @ K_bkv^T [dk,S] -> attn (raw, pre-softmax)
  //    z = b*H + h ; A inner stride = h*128 ; B inner stride = (h/4)*128
  wmma_gemm_f32<true><<<dim3(S / BN, S / BM, B * H), blk, 0, stream>>>(
      Qp, Kp, attn, S, S, DK, D, KV * DK, S,
      H,
      (long long)S * D, DK, 1,                 // A: Q
      (long long)S * (KV * DK), DK, 4,         // B: K (grouped kv heads)
      (long long)H * S * S, (long long)S * S, 1);

  // 5) softmax rows (applies scale), in place on attn
  softmax_rows_2048<<<dim3(B * H * S), blk, 0, stream>>>(attn, scale);

  // 6) ctx[b,h] = attn_bh [S,S] @ V_bkv [S,dk] -> Cx (packed [B*S, D], col h*128)
  wmma_gemm_f32<false><<<dim3(DK / BN, S / BM, B * H), blk, 0, stream>>>(
      attn, Vp, Cx, S, DK, S, S, KV * DK, D,
      H,
      (long long)H * S * S, (long long)S * S, 1,   // A: attn
      (long long)S * (KV * DK), DK, 4,             // B: V (grouped kv heads)
      (long long)S * D, DK, 1);                    // C: ctx column block h*128

  // 7) out = ctx @ Wo^T
  wmma_gemm_f32<true><<<dim3(D / BN, BS / BM, 1), blk, 0, stream>>>(
      Cx, Wo, out, (int)BS, D, D, D, D, D, 1, 0, 0, 1, 0, 0, 1, 0, 0, 1);
}